// SimulatedMaxPool3D_p1_34402688041511
// MI455X (gfx1250) — compile-verified
//
#include <hip/hip_runtime.h>

// SimulatedMaxPool3D on MI455X (gfx1250): 2x2/stride-2 max over (H,W) of
// (1,32,64,256,256) fp32. Pure HBM-streaming problem (~640 MiB traffic,
// ~27 us floor at 23.3 TB/s). No matrix math => no WMMA; optimize the
// data-movement path instead: b128 loads/stores with non-temporal hints
// (input stream is 512 MiB > 192 MB L2, zero reuse) + global_prefetch_b8.

typedef float v4f __attribute__((ext_vector_type(4)));

#define CH   32
#define DEP  64
#define HIN  256
#define WIN  256
#define HOUT 128
#define WOUT 128

// total outputs = CH*DEP*HOUT*WOUT = 33,554,432 ; /4 = 8,388,608 float4s
#define NVEC4 (CH * DEP * HOUT * WOUT / 4)

__global__ __launch_bounds__(256)
void maxpool2x2_nt_kernel(const float* __restrict__ x,
                          float* __restrict__ out,
                          long total_out_elems) {
    const long v = (long)blockIdx.x * blockDim.x + threadIdx.x; // float4 index
    if (v < (long)NVEC4) {
        // v -> (plane = c*DEP+d, oh, col4)
        const int  col4  = (int)(v & 31);        // WOUT/4 = 32 vec4 per row
        const long row   = v >> 5;               // global output row
        const int  oh    = (int)(row & (HOUT - 1));
        const long plane = row >> 7;             // row / HOUT

        const float* src = x + plane * (long)(HIN * WIN)
                             + (long)(2 * oh) * WIN
                             + (long)col4 * 8;   // 8 input columns per thread

        // Speculative prefetch ~16 input rows ahead of this thread's stream
        // (gfx1250 global_prefetch_b8; silently dropped if OOB).
        __builtin_prefetch(src + 16 * WIN, 0, 1);

        const v4f* p0 = (const v4f*)(src);            // row 2*oh,   cols [8c..8c+3]
        const v4f* p1 = (const v4f*)(src + 4);        // row 2*oh,   cols [8c+4..8c+7]
        const v4f* p2 = (const v4f*)(src + WIN);      // row 2*oh+1, cols [8c..8c+3]
        const v4f* p3 = (const v4f*)(src + WIN + 4);  // row 2*oh+1, cols [8c+4..8c+7]

        // Non-temporal b128 loads: streamed once, never re-read.
        v4f a0 = __builtin_nontemporal_load(p0);
        v4f a1 = __builtin_nontemporal_load(p1);
        v4f b0 = __builtin_nontemporal_load(p2);
        v4f b1 = __builtin_nontemporal_load(p3);

        v4f o;
        o.x = fmaxf(fmaxf(a0.x, a0.y), fmaxf(b0.x, b0.y));
        o.y = fmaxf(fmaxf(a0.z, a0.w), fmaxf(b0.z, b0.w));
        o.z = fmaxf(fmaxf(a1.x, a1.y), fmaxf(b1.x, b1.y));
        o.w = fmaxf(fmaxf(a1.z, a1.w), fmaxf(b1.z, b1.w));

        // Non-temporal b128 store: output not consumed by this dispatch.
        __builtin_nontemporal_store(o, (v4f*)(out + v * 4));
    }

    // Second tuple element: h = x.shape[3] = 256, appended after the pooled
    // tensor in d_out (tuple outputs are concatenated flat, float dtype).
    if (v == 0) {
        out[total_out_elems - 1] = 256.0f;
    }
}

extern "C" void kernel_launch(void* const* d_in, const int* in_sizes, int n_in,
                              void* d_out, int out_size, void* d_ws, size_t ws_size,
                              hipStream_t stream) {
    (void)in_sizes; (void)n_in; (void)d_ws; (void)ws_size;
    const float* x  = (const float*)d_in[0];
    float* out      = (float*)d_out;

    const int  block = 256;                       // 8 wave32s
    const long grid  = (NVEC4 + block - 1) / block; // 32768 blocks, exact cover

    maxpool2x2_nt_kernel<<<(dim3)(unsigned)grid, block, 0, stream>>>(
        x, out, (long)out_size);
}